// ALIGNNLayer_46102178955277
// MI455X (gfx1250) — compile-verified
//
#include <hip/hip_runtime.h>

typedef __attribute__((ext_vector_type(2))) float v2f;
typedef __attribute__((ext_vector_type(8))) float v8f;

#define DIM 64
#define LDS_STRIDE 66   // pad: keeps float2 A-frag reads 8B-aligned, breaks bank conflicts

// ---------------------------------------------------------------------------
// zero fill (graph-capture-safe replacement for memset)
// ---------------------------------------------------------------------------
__global__ void zero_f32(float* __restrict__ p, long n) {
  long i = (long)blockIdx.x * blockDim.x + threadIdx.x;
  if (i < n) p[i] = 0.0f;
}

// ---------------------------------------------------------------------------
// Multi-output tile GEMM:  out[s] = X @ W[widx_s] + b[widx_s]   (K = N = 64)
// blockDim.x = nslots*128 (4 waves per slot, one 16-col tile each).
// Each block handles one 16-row tile of X (grid-stride).
// Uses V_WMMA_F32_16X16X4_F32: full fp32 precision on the matrix pipe.
// ---------------------------------------------------------------------------
__global__ void __launch_bounds__(512) gemm_ws(
    const float* __restrict__ X, const float* __restrict__ W, const float* __restrict__ B,
    float* __restrict__ o0, float* __restrict__ o1,
    float* __restrict__ o2, float* __restrict__ o3,
    int i0, int i1, int i2, int i3, int ntiles)
{
  __shared__ float lds[16 * LDS_STRIDE];

  const int tid  = threadIdx.x;
  const int wave = tid >> 5;
  const int lane = tid & 31;
  const int slot = wave >> 2;       // which weight matrix
  const int ct   = wave & 3;        // which 16-column tile of the 64 outputs

  const int widx = (slot == 0) ? i0 : (slot == 1) ? i1 : (slot == 2) ? i2 : i3;
  float* out     = (slot == 0) ? o0 : (slot == 1) ? o1 : (slot == 2) ? o2 : o3;

  // WMMA 16x16x4 f32 layouts (ISA 7.12.2):
  //  A (16x4):  lanes 0-15 hold M=lane {K=k0,k0+1}; lanes 16-31 hold {K=k0+2,k0+3}
  //  B (4x16):  lane gives N=lane&15; VGPR0/1 hold K=k0+koff / k0+koff+1
  //  C/D:       VGPR r -> M = r + (lane<16 ? 0 : 8), N = lane&15
  const int n    = ct * 16 + (lane & 15);
  const int koff = (lane >> 4) << 1;   // 0 for lanes 0-15, 2 for lanes 16-31
  const int m    = lane & 15;

  // Preload B fragments for the whole K=64 reduction (16 steps of K=4).
  // W is 5*64*64 f32 = 80KB -> fully L2-resident, broadcast across waves.
  v2f bf[16];
  #pragma unroll
  for (int ks = 0; ks < 16; ++ks) {
    int k = ks * 4 + koff;
    bf[ks].x = W[((widx * DIM + k    ) * DIM) + n];
    bf[ks].y = W[((widx * DIM + k + 1) * DIM) + n];
  }
  const float bias = B[widx * DIM + n];

  for (int t = blockIdx.x; t < ntiles; t += gridDim.x) {
    __syncthreads();
    // cooperative stage of the 16x64 A tile into LDS
    for (int idx = tid; idx < 16 * DIM; idx += blockDim.x) {
      int r = idx >> 6, c = idx & 63;
      lds[r * LDS_STRIDE + c] = X[((size_t)t * 16 + r) * DIM + c];
    }
    __syncthreads();

    v8f acc;
    #pragma unroll
    for (int r = 0; r < 8; ++r) acc[r] = bias;   // fold bias into accumulator

    #pragma unroll
    for (int ks = 0; ks < 16; ++ks) {
      v2f a = *(const v2f*)&lds[m * LDS_STRIDE + ks * 4 + koff];
      acc = __builtin_amdgcn_wmma_f32_16x16x4_f32(
          /*neg_a=*/false, a, /*neg_b=*/false, bf[ks],
          /*c_mod=*/(short)0, acc, /*reuse_a=*/false, /*reuse_b=*/false);
    }

    const size_t row0 = (size_t)t * 16 + ((lane >> 4) << 3);
    #pragma unroll
    for (int r = 0; r < 8; ++r)
      out[(row0 + r) * DIM + n] = acc[r];
  }
}

// ---------------------------------------------------------------------------
// Edge kernel: e_hat = Dh[dst] + Eh[src] + Ce ; sigma = sigmoid(e_hat)
//   eh_out   = relu(e_hat)                     (edge output of the layer)
//   num[dst] += sigma * Bh[src], den[dst] += sigma   (f32 atomics)
// One wave per edge, 2 columns per lane (float2 -> coalesced 256B rows).
// ---------------------------------------------------------------------------
__global__ void __launch_bounds__(256) edge_gate(
    const float* __restrict__ Dh, const float* __restrict__ Eh,
    const float* __restrict__ Ce, const float* __restrict__ Bh,
    const int* __restrict__ src, const int* __restrict__ dst, long nE,
    float* __restrict__ eh_out, float* __restrict__ num, float* __restrict__ den)
{
  long gid  = (long)blockIdx.x * blockDim.x + threadIdx.x;
  long edge = gid >> 5;
  int  lane = threadIdx.x & 31;
  if (edge >= nE) return;

  int s = src[edge];   // wave-uniform -> scalar/broadcast load
  int d = dst[edge];
  int c = lane * 2;

  v2f dh = *(const v2f*)(Dh + (size_t)d    * DIM + c);
  v2f eh = *(const v2f*)(Eh + (size_t)s    * DIM + c);
  v2f ce = *(const v2f*)(Ce + (size_t)edge * DIM + c);

  v2f ehat = dh + eh + ce;
  v2f sig, rel;
  sig.x = 1.0f / (1.0f + __expf(-ehat.x));
  sig.y = 1.0f / (1.0f + __expf(-ehat.y));
  rel.x = fmaxf(ehat.x, 0.0f);
  rel.y = fmaxf(ehat.y, 0.0f);

  *(v2f*)(eh_out + (size_t)edge * DIM + c) = rel;

  v2f bh = *(const v2f*)(Bh + (size_t)s * DIM + c);
  atomicAdd(&num[(size_t)d * DIM + c    ], sig.x * bh.x);
  atomicAdd(&num[(size_t)d * DIM + c + 1], sig.y * bh.y);
  atomicAdd(&den[(size_t)d * DIM + c    ], sig.x);
  atomicAdd(&den[(size_t)d * DIM + c + 1], sig.y);
}

// ---------------------------------------------------------------------------
// Node update: out = relu(Ah + num / (den + eps))
// ---------------------------------------------------------------------------
__global__ void __launch_bounds__(256) node_update(
    const float* __restrict__ Ah, const float* __restrict__ num,
    const float* __restrict__ den, float* __restrict__ out, long n)
{
  long i = (long)blockIdx.x * blockDim.x + threadIdx.x;
  if (i < n) out[i] = fmaxf(Ah[i] + num[i] / (den[i] + 1e-6f), 0.0f);
}

// ---------------------------------------------------------------------------
extern "C" void kernel_launch(void* const* d_in, const int* in_sizes, int n_in,
                              void* d_out, int out_size, void* d_ws, size_t ws_size,
                              hipStream_t stream) {
  (void)n_in; (void)out_size; (void)ws_size;

  const float* h      = (const float*)d_in[0];
  const float* e      = (const float*)d_in[1];
  const float* l      = (const float*)d_in[2];
  const int*   src    = (const int*)  d_in[3];
  const int*   dst    = (const int*)  d_in[4];
  const int*   lg_src = (const int*)  d_in[5];
  const int*   lg_dst = (const int*)  d_in[6];
  const float* Wn     = (const float*)d_in[7];
  const float* bn     = (const float*)d_in[8];
  const float* We     = (const float*)d_in[9];
  const float* be     = (const float*)d_in[10];

  const long nNodes = in_sizes[0] / DIM;   // 100000
  const long nEdges = in_sizes[1] / DIM;   // 1000000
  const long nLg    = in_sizes[2] / DIM;   // 2000000
  const size_t NH = (size_t)nNodes * DIM;
  const size_t NE = (size_t)nEdges * DIM;
  const size_t NL = (size_t)nLg    * DIM;

  // ---- workspace layout (floats); layer-2 aliases dead layer-1 buffers ----
  float* ws   = (float*)d_ws;
  float* m_   = ws;                 // [0, NE)      : relu edge output of layer1
  float* Ah   = ws + NE;            // layer1 node linears
  float* Bh   = Ah + NH;
  float* Dh   = Bh + NH;
  float* Eh   = Dh + NH;
  float* num1 = Eh + NH;
  float* den1 = num1 + NH;          // num1,den1 contiguous -> single zero pass
  float* Ce   = den1 + NH;          // [NE+6NH, 2NE+6NH)
  // layer2 (reuse [NE, ...) once h1 is written; [5NE,7NE)+Cl are fresh space)
  float* Am   = ws + NE;
  float* Bm   = Am + NE;
  float* Dm   = Bm + NE;
  float* Em   = Dm + NE;
  float* num2 = Em + NE;
  float* den2 = num2 + NE;          // contiguous with num2
  float* Cl   = den2 + NE;          // [7NE, 7NE+NL)

  float* h1 = (float*)d_out;        // [NH]
  float* e1 = h1 + NH;              // [NE]
  float* l1 = e1 + NE;              // [NL]

  const int ntilesH = (int)(nNodes / 16);  // 6250
  const int ntilesE = (int)(nEdges / 16);  // 62500
  const int ntilesL = (int)(nLg    / 16);  // 125000

  // zero accumulators (regions are disjoint from all live layer-1/2 writes)
  {
    long n1 = 2 * (long)NH;
    zero_f32<<<(unsigned)((n1 + 255) / 256), 256, 0, stream>>>(num1, n1);
    long n2 = 2 * (long)NE;
    zero_f32<<<(unsigned)((n2 + 255) / 256), 256, 0, stream>>>(num2, n2);
  }

  // -------- layer 1: gated_gcn(h, e, src, dst, Wn, bn) --------
  // Ah,Bh,Dh,Eh = h @ Wn[{0,1,3,4}] + bn
  gemm_ws<<<ntilesH, 512, 0, stream>>>(h, Wn, bn, Ah, Bh, Dh, Eh,
                                       0, 1, 3, 4, ntilesH);
  // Ce = e @ Wn[2] + bn[2]
  gemm_ws<<<ntilesE, 128, 0, stream>>>(e, Wn, bn, Ce, nullptr, nullptr, nullptr,
                                       2, 2, 2, 2, ntilesE);
  // edge gate + segment sums; m_ = relu(e_hat)
  {
    long thr = nEdges * 32;
    edge_gate<<<(unsigned)(thr / 256), 256, 0, stream>>>(
        Dh, Eh, Ce, Bh, src, dst, nEdges, m_, num1, den1);
  }
  // h1 = relu(Ah + num1/(den1+eps))
  node_update<<<(unsigned)((NH + 255) / 256), 256, 0, stream>>>(
      Ah, num1, den1, h1, (long)NH);

  // -------- layer 2: gated_gcn(m_, l, lg_src, lg_dst, We, be) --------
  gemm_ws<<<ntilesE, 512, 0, stream>>>(m_, We, be, Am, Bm, Dm, Em,
                                       0, 1, 3, 4, ntilesE);
  gemm_ws<<<ntilesL, 128, 0, stream>>>(l, We, be, Cl, nullptr, nullptr, nullptr,
                                       2, 2, 2, 2, ntilesL);
  {
    long thr = nLg * 32;
    edge_gate<<<(unsigned)(thr / 256), 256, 0, stream>>>(
        Dm, Em, Cl, Bm, lg_src, lg_dst, nLg, l1, num2, den2);
  }
  // e1 = relu(Am + num2/(den2+eps))
  node_update<<<(unsigned)((NE + 255) / 256), 256, 0, stream>>>(
      Am, num2, den2, e1, (long)NE);
}